// MyModel_81423989998055
// MI455X (gfx1250) — compile-verified
//
#include <hip/hip_runtime.h>
#include <math.h>

// ---------------------------------------------------------------------------
// Canny pipeline for MI455X (gfx1250), bandwidth-bound (~17us roofline).
// K1: img -> blurred, grad_x, grad_y, mag, orientation   (fused, LDS tiles)
// K2: mag+ori -> NMS + double threshold -> u8 code map in workspace
//     (directional differences computed with V_WMMA_F32_16X16X4_F32)
// K3: code map -> hysteresis -> thin_edges
// Halo tiles loaded with global_load_async_to_lds_b32 (ASYNCcnt path) on
// interior blocks; boundary blocks use a zero-padded fallback.
// ---------------------------------------------------------------------------

typedef __attribute__((ext_vector_type(2))) float v2f;
typedef __attribute__((ext_vector_type(8))) float v8f;

namespace {
constexpr int HH = 1024, WW = 1024, BB = 8, CC = 3;
constexpr long long PL = (long long)HH * WW;              // 1048576
constexpr long long BLUR_OFF = 0;
constexpr long long GX_OFF   = (long long)BB * CC * PL;   // 25165824
constexpr long long GY_OFF   = GX_OFF + (long long)BB * PL;
constexpr long long MAG_OFF  = GY_OFF + (long long)BB * PL;
constexpr long long ORI_OFF  = MAG_OFF + (long long)BB * PL;
constexpr long long THIN_OFF = ORI_OFF + (long long)BB * PL;

// normalized gaussian(3x3, sigma=1) weights
constexpr float WCOR = 0.07511360795411207f;
constexpr float WEDG = 0.12384140878046110f;
constexpr float WCEN = 0.20418012544050554f;

constexpr float PI_F = 3.14159265358979323846f;

// 12 taps: t0 = center(+1 for every filter), t1..t8 = 8 directional
// neighbors (filter f has -1 at tap f+1), t9..t11 = zero-weight padding.
// offs order: (0,1),(-1,1),(-1,0),(-1,-1),(0,-1),(1,-1),(1,0),(1,1)
// packed (d+1) two bits per tap:
constexpr unsigned DYP =
    (1u<<0)|(1u<<2)|(0u<<4)|(0u<<6)|(0u<<8)|(1u<<10)|(2u<<12)|(2u<<14)|
    (2u<<16)|(1u<<18)|(1u<<20)|(1u<<22);
constexpr unsigned DXP =
    (1u<<0)|(2u<<2)|(2u<<4)|(1u<<6)|(0u<<8)|(0u<<10)|(0u<<12)|(1u<<14)|
    (2u<<16)|(1u<<18)|(1u<<20)|(1u<<22);
} // namespace

__device__ __forceinline__ int tap_dy(int t) { return (int)((DYP >> (2*t)) & 3u) - 1; }
__device__ __forceinline__ int tap_dx(int t) { return (int)((DXP >> (2*t)) & 3u) - 1; }

// ---- CDNA5 async global->LDS copy (ASYNCcnt) ------------------------------
__device__ __forceinline__ void async_ld_b32(const void* gptr, void* lptr) {
    unsigned loff = (unsigned)(unsigned long long)lptr;   // LDS byte offset = addr[31:0]
    asm volatile("global_load_async_to_lds_b32 %0, %1, off"
                 :: "v"(loff), "v"(gptr) : "memory");
}
__device__ __forceinline__ void async_wait0() {
    asm volatile("s_wait_asynccnt 0" ::: "memory");
}

// ---------------------------------------------------------------------------
// K1: blur + sobel + magnitude + orientation. 16x16 tile, img halo 2.
// ---------------------------------------------------------------------------
__global__ __launch_bounds__(256)
void canny_stage1(const float* __restrict__ img, float* __restrict__ out)
{
    __shared__ float imgL[CC * 20 * 20];   // 1200 floats, halo-2 img tile
    __shared__ float blrL[CC * 18 * 18];   //  972 floats, halo-1 blurred

    const int tid = threadIdx.x;
    const int tx0 = blockIdx.x * 16, ty0 = blockIdx.y * 16;
    const int b   = blockIdx.z;

    const bool interior = (tx0 >= 2) && (ty0 >= 2) &&
                          (tx0 + 18 <= WW) && (ty0 + 18 <= HH);

    if (interior) {
        // async fast path: 1200 B32 copies, 5 rounds of 256 lanes
        #pragma unroll
        for (int i = 0; i < 5; ++i) {
            int idx = tid + i * 256;
            if (idx < CC * 400) {
                int ch = idx / 400, rem = idx - ch * 400;
                int ly = rem / 20, lx = rem - ly * 20;
                const float* g = img + (size_t)(b * CC + ch) * PL
                                     + (size_t)(ty0 - 2 + ly) * WW + (tx0 - 2 + lx);
                async_ld_b32(g, &imgL[idx]);
            }
        }
        async_wait0();
        __syncthreads();
    } else {
        // boundary: zero-padded scalar loads
        for (int i = 0; i < 5; ++i) {
            int idx = tid + i * 256;
            if (idx < CC * 400) {
                int ch = idx / 400, rem = idx - ch * 400;
                int ly = rem / 20, lx = rem - ly * 20;
                int gy = ty0 - 2 + ly, gx = tx0 - 2 + lx;
                float v = 0.0f;
                if (gy >= 0 && gy < HH && gx >= 0 && gx < WW)
                    v = img[(size_t)(b * CC + ch) * PL + (size_t)gy * WW + gx];
                imgL[idx] = v;
            }
        }
        __syncthreads();
    }

    // gaussian blur at tile + halo-1 (3 * 18 * 18 = 972 positions)
    for (int p = tid; p < CC * 324; p += 256) {
        int ch = p / 324, rem = p - ch * 324;
        int ly = rem / 18, lx = rem - ly * 18;
        const float* s = &imgL[ch * 400 + ly * 20 + lx];   // 3x3 window top-left
        float v = WCOR * (s[0] + s[2] + s[40] + s[42])
                + WEDG * (s[1] + s[20] + s[22] + s[41])
                + WCEN * s[21];
        blrL[p] = v;
    }
    __syncthreads();

    // per-thread center pixel: sobel summed over channels
    const int y = tid >> 4, x = tid & 15;
    const size_t opix = (size_t)(ty0 + y) * WW + (tx0 + x);
    float gxs = 0.0f, gys = 0.0f;
    #pragma unroll
    for (int ch = 0; ch < CC; ++ch) {
        const float* s = &blrL[ch * 324 + y * 18 + x];     // window rows y..y+2
        float b00 = s[0],  b01 = s[1],  b02 = s[2];
        float b10 = s[18], b11 = s[19], b12 = s[20];
        float b20 = s[36], b21 = s[37], b22 = s[38];
        __builtin_nontemporal_store(b11, out + BLUR_OFF + (size_t)(b * CC + ch) * PL + opix);
        // w_sx = [[-.5,0,.5],[-1,0,1],[-.5,0,.5]], w_sy = transpose
        gxs += 0.5f * (b02 - b00) + (b12 - b10) + 0.5f * (b22 - b20);
        gys += 0.5f * (b20 - b00) + (b21 - b01) + 0.5f * (b22 - b02);
    }
    const float gxv = gxs * (1.0f / 3.0f);
    const float gyv = gys * (1.0f / 3.0f);
    const float mag = sqrtf(gxv * gxv + gyv * gyv);
    // arctan(gy/gx)*(360/pi)+180, rounded to multiple of 45 (half-even = rint)
    const float ori = rintf((atanf(gyv / gxv) * (360.0f / PI_F) + 180.0f)
                            * (1.0f / 45.0f)) * 45.0f;

    __builtin_nontemporal_store(gxv, out + GX_OFF + (size_t)b * PL + opix);
    __builtin_nontemporal_store(gyv, out + GY_OFF + (size_t)b * PL + opix);
    out[MAG_OFF + (size_t)b * PL + opix] = mag;   // re-read by K2: keep RT
    out[ORI_OFF + (size_t)b * PL + opix] = ori;   // re-read by K2: keep RT
}

// ---------------------------------------------------------------------------
// K2: NMS via WMMA (16 filters x 12 taps x 16 pixels) + double threshold.
// Writes {0,1,2} u8 code per pixel to workspace.
// ---------------------------------------------------------------------------
__device__ __forceinline__ float wA(int m, int t) {
    // filter m (<8): +1 at tap0 (center), -1 at tap m+1, else 0; filters 8..15: 0
    return (m < 8) ? ((t == 0) ? 1.0f : ((t == m + 1) ? -1.0f : 0.0f)) : 0.0f;
}

__global__ __launch_bounds__(256)
void canny_stage2(const float* __restrict__ out, unsigned char* __restrict__ enc)
{
    __shared__ float magL[18 * 18];   // mag tile with halo 1

    const int tid = threadIdx.x;
    const int tx0 = blockIdx.x * 16, ty0 = blockIdx.y * 16;
    const int b   = blockIdx.z;
    const float* magp = out + MAG_OFF + (size_t)b * PL;

    const bool interior = (tx0 >= 1) && (ty0 >= 1) &&
                          (tx0 + 17 <= WW) && (ty0 + 17 <= HH);

    if (interior) {
        #pragma unroll
        for (int i = 0; i < 2; ++i) {
            int idx = tid + i * 256;
            if (idx < 324) {
                int ly = idx / 18, lx = idx - ly * 18;
                const float* g = magp + (size_t)(ty0 - 1 + ly) * WW + (tx0 - 1 + lx);
                async_ld_b32(g, &magL[idx]);
            }
        }
        async_wait0();
        __syncthreads();
    } else {
        for (int i = 0; i < 2; ++i) {
            int idx = tid + i * 256;
            if (idx < 324) {
                int ly = idx / 18, lx = idx - ly * 18;
                int gy = ty0 - 1 + ly, gx = tx0 - 1 + lx;
                float v = 0.0f;
                if (gy >= 0 && gy < HH && gx >= 0 && gx < WW)
                    v = magp[(size_t)gy * WW + gx];
                magL[idx] = v;
            }
        }
        __syncthreads();
    }

    const int lane  = tid & 31;
    const int wave  = tid >> 5;        // 8 waves, 2 rows each
    const int n     = lane & 15;       // pixel column within 16-wide strip
    const int m     = lane & 15;       // filter row of A this lane holds
    const int khalf = (lane < 16) ? 0 : 2;   // K pair handled by this half-wave

    #pragma unroll
    for (int rr = 0; rr < 2; ++rr) {
        const int r = 2 * wave + rr;
        v8f acc = {0.f, 0.f, 0.f, 0.f, 0.f, 0.f, 0.f, 0.f};
        #pragma unroll
        for (int c3 = 0; c3 < 3; ++c3) {
            const int t0 = 4 * c3 + khalf, t1 = t0 + 1;
            // A: 16x4 f32 — lanes 0-15 hold K={0,1}, lanes 16-31 hold K={2,3}
            v2f av = { wA(m, t0), wA(m, t1) };
            // B: 4x16 f32 im2col — shifted LDS reads (zero halo handles pad)
            v2f bv = { magL[(1 + r + tap_dy(t0)) * 18 + (1 + n + tap_dx(t0))],
                       magL[(1 + r + tap_dy(t1)) * 18 + (1 + n + tap_dx(t1))] };
            acc = __builtin_amdgcn_wmma_f32_16x16x4_f32(
                      false, av, false, bv, (short)0, acc, false, false);
        }
        if (lane < 16) {
            // lane n holds directional[0..7] for pixel (r, n) in acc[0..7]
            const float magc = magL[(1 + r) * 18 + (1 + n)];
            const float ori  = out[ORI_OFF + (size_t)b * PL
                                   + (size_t)(ty0 + r) * WW + (tx0 + n)];
            // positive_idx = (ori/45) % 8: ori/45 is an exact integer in
            // {0..8} (or NaN); modulo just maps 8 -> 0. NaN stays NaN so all
            // equality tests below are false, exactly like the reference.
            const float q    = ori * (1.0f / 45.0f);
            const float pidx = (q == 8.0f) ? 0.0f : q;
            float thin = magc;
            #pragma unroll
            for (int pos = 0; pos < 4; ++pos) {
                bool is_or  = (pidx == (float)pos) || (pidx == (float)(pos + 4));
                bool is_max = fminf(acc[pos], acc[pos + 4]) > 0.0f;
                if (is_or && !is_max) thin = 0.0f;
            }
            unsigned char e = (thin > 0.3f) ? 2 : ((thin > 0.15f) ? 1 : 0);
            enc[(size_t)b * PL + (size_t)(ty0 + r) * WW + (tx0 + n)] = e;
        }
    }
}

// ---------------------------------------------------------------------------
// K3: hysteresis on the u8 code map. conv(t,1.25*ones)>1  <=>  sum(enc) >= 2.
// ---------------------------------------------------------------------------
__global__ __launch_bounds__(256)
void canny_stage3(const unsigned char* __restrict__ enc, float* __restrict__ out)
{
    const int tid = threadIdx.x;
    const int gx = blockIdx.x * 16 + (tid & 15);
    const int gy = blockIdx.y * 16 + (tid >> 4);
    const int b  = blockIdx.z;
    const unsigned char* ep = enc + (size_t)b * PL;

    const unsigned char e = ep[(size_t)gy * WW + gx];
    float v = 0.0f;
    if (e == 2) {
        v = 1.0f;                       // high: 1 + 0 (weak=0)
    } else if (e == 1) {                // weak: promote if 3x3 sum >= 2
        int s = 0;
        #pragma unroll
        for (int dy = -1; dy <= 1; ++dy)
            #pragma unroll
            for (int dx = -1; dx <= 1; ++dx) {
                int yy = gy + dy, xx = gx + dx;
                if (yy >= 0 && yy < HH && xx >= 0 && xx < WW)
                    s += (int)ep[(size_t)yy * WW + xx];
            }
        if (s >= 2) v = 1.0f;
    }
    __builtin_nontemporal_store(v, out + THIN_OFF + (size_t)b * PL
                                   + (size_t)gy * WW + gx);
}

// ---------------------------------------------------------------------------
extern "C" void kernel_launch(void* const* d_in, const int* in_sizes, int n_in,
                              void* d_out, int out_size, void* d_ws, size_t ws_size,
                              hipStream_t stream)
{
    (void)in_sizes; (void)n_in; (void)out_size; (void)ws_size;
    const float* img = (const float*)d_in[0];   // conv weights d_in[1..5] baked in
    float* out = (float*)d_out;
    unsigned char* enc = (unsigned char*)d_ws;  // needs B*H*W = 8 MiB of scratch

    dim3 grid(WW / 16, HH / 16, BB), block(256);
    canny_stage1<<<grid, block, 0, stream>>>(img, out);
    canny_stage2<<<grid, block, 0, stream>>>(out, enc);
    canny_stage3<<<grid, block, 0, stream>>>(enc, out);
}